// vMF_22170621182543
// MI455X (gfx1250) — compile-verified
//
#include <hip/hip_runtime.h>

typedef __attribute__((ext_vector_type(2)))  float    v2f;
typedef __attribute__((ext_vector_type(8)))  float    v8f;

#define B_DIM 64
#define T_DIM 512
#define M_DIM 512
#define PAD_IDX 1
#define LAMBDA1 0.02f
#define LAMBDA2 0.1f

__device__ __forceinline__ float wave_reduce_sum(float v) {
#pragma unroll
  for (int m = 16; m > 0; m >>= 1) v += __shfl_xor(v, m, 32);
  return v;
}

// One block per t. 8 wave32s; wave w owns rows b = w, w+8, ..., w+56.
// Per-wave partial vectors live in REGISTERS (16 M-slots per lane per matrix)
// and are spilled to LDS once after the streaming loop.
__global__ void __launch_bounds__(256)
vmf_per_t_kernel(const float* __restrict__ outputs,   // [B,T,M]
                 const int*   __restrict__ targets,   // [B,T]
                 const float* __restrict__ embedding, // [V,M]
                 float* __restrict__ partial) {       // [T]
  const int t    = blockIdx.x;
  const int tid  = threadIdx.x;
  const int wave = tid >> 5;
  const int lane = tid & 31;

  __shared__ float ps_out[8][M_DIM];   // per-wave partial sums (atomic-free)
  __shared__ float ps_trg[8][M_DIM];
  __shared__ float s_out[M_DIM];
  __shared__ float s_trg[M_DIM];
  __shared__ float wA[8];
  __shared__ int   wN[8];

  // register accumulators: lane owns M-indices lane*4 + j*128 + c
  float aco[16];
  float act[16];
#pragma unroll
  for (int i = 0; i < 16; ++i) { aco[i] = 0.0f; act[i] = 0.0f; }

  float scalarAcc = 0.0f;
  int   nValid    = 0;

  for (int b = wave; b < B_DIM; b += 8) {
    const float* orow = outputs + ((size_t)b * T_DIM + t) * M_DIM;
    int tgt = targets[b * T_DIM + t];
    const float* erow = embedding + (size_t)tgt * M_DIM;

    // streaming prefetch of next owned rows -> global_prefetch_b8
    if (b + 8 < B_DIM) {
      __builtin_prefetch(outputs + ((size_t)(b + 8) * T_DIM + t) * M_DIM + lane * 4, 0, 0);
      int tgt_n = targets[(b + 8) * T_DIM + t];
      __builtin_prefetch(embedding + (size_t)tgt_n * M_DIM + lane * 4, 0, 0);
    }

    // ---- outputs row: norm + normalized accumulate (registers) ----
    float4 ov[4];
    float sq = 0.0f;
#pragma unroll
    for (int j = 0; j < 4; ++j) {
      ov[j] = *reinterpret_cast<const float4*>(orow + lane * 4 + j * 128);
      sq += ov[j].x * ov[j].x + ov[j].y * ov[j].y + ov[j].z * ov[j].z + ov[j].w * ov[j].w;
    }
    float kappa2 = wave_reduce_sum(sq);
    float kappa  = sqrtf(kappa2);
    // logC_m(kappa), m=512 -> v=255: sqrt(256^2+z^2) - 254*log(254 + sqrt(254^2+z^2))
    float logc = sqrtf(65536.0f + kappa2) - 254.0f * logf(254.0f + sqrtf(64516.0f + kappa2));
    scalarAcc += (-logc + LAMBDA1 * kappa);
    float oinv = 1.0f / fmaxf(kappa, 1e-12f);
#pragma unroll
    for (int j = 0; j < 4; ++j) {
      aco[j * 4 + 0] += ov[j].x * oinv;
      aco[j * 4 + 1] += ov[j].y * oinv;
      aco[j * 4 + 2] += ov[j].z * oinv;
      aco[j * 4 + 3] += ov[j].w * oinv;
    }

    // ---- embedding gather row: norm + masked normalized accumulate ----
    float4 ev[4];
    float esq = 0.0f;
#pragma unroll
    for (int j = 0; j < 4; ++j) {
      ev[j] = *reinterpret_cast<const float4*>(erow + lane * 4 + j * 128);
      esq += ev[j].x * ev[j].x + ev[j].y * ev[j].y + ev[j].z * ev[j].z + ev[j].w * ev[j].w;
    }
    float en2   = wave_reduce_sum(esq);
    float maskf = (tgt != PAD_IDX) ? 1.0f : 0.0f;
    nValid += (tgt != PAD_IDX) ? 1 : 0;
    float einv = maskf / fmaxf(sqrtf(en2), 1e-12f);
#pragma unroll
    for (int j = 0; j < 4; ++j) {
      act[j * 4 + 0] += ev[j].x * einv;
      act[j * 4 + 1] += ev[j].y * einv;
      act[j * 4 + 2] += ev[j].z * einv;
      act[j * 4 + 3] += ev[j].w * einv;
    }
  }

  // one-shot spill of register partials to LDS (ds_store_b128 x4 per matrix)
#pragma unroll
  for (int j = 0; j < 4; ++j) {
    float4 vo = make_float4(aco[j * 4 + 0], aco[j * 4 + 1], aco[j * 4 + 2], aco[j * 4 + 3]);
    float4 vt = make_float4(act[j * 4 + 0], act[j * 4 + 1], act[j * 4 + 2], act[j * 4 + 3]);
    *reinterpret_cast<float4*>(&ps_out[wave][lane * 4 + j * 128]) = vo;
    *reinterpret_cast<float4*>(&ps_trg[wave][lane * 4 + j * 128]) = vt;
  }
  if (lane == 0) { wA[wave] = scalarAcc; wN[wave] = nValid; }
  __syncthreads();

  // combine the 8 per-wave partial vectors
  for (int m = tid; m < M_DIM; m += 256) {
    float so = 0.0f, st = 0.0f;
#pragma unroll
    for (int w = 0; w < 8; ++w) { so += ps_out[w][m]; st += ps_trg[w][m]; }
    s_out[m] = so;
    s_trg[m] = st;
  }
  __syncthreads();

  // wave 0 (uniform EXEC across its 32 lanes) finishes the per-t result
  if (wave == 0) {
    float termA = 0.0f;
    int   nv    = 0;
#pragma unroll
    for (int w = 0; w < 8; ++w) { termA += wA[w]; nv += wN[w]; }

    // f32 WMMA diagonal dot: A[i,k]=s_out[4i+k], B[k,i]=s_trg[4i+k]
    // => C[i,i] accumulates chunk-i dot products; 8 issues cover 512 elems.
    // A 16x4 f32 layout: lanes<16 hold K=0(v0),1(v1); lanes>=16 hold K=2,3.
    const int idx4 = lane & 15;
    const int koff = (lane < 16) ? 0 : 2;
    v8f acc = {};
#pragma unroll
    for (int base = 0; base < M_DIM; base += 64) {
      v2f a, bb;
      a.x  = s_out[base + idx4 * 4 + koff];
      a.y  = s_out[base + idx4 * 4 + koff + 1];
      bb.x = s_trg[base + idx4 * 4 + koff];
      bb.y = s_trg[base + idx4 * 4 + koff + 1];
      acc = __builtin_amdgcn_wmma_f32_16x16x4_f32(
          false, a, false, bb, (short)0, acc, false, false);
    }
    float dot = 0.0f;
#pragma unroll
    for (int r = 0; r < 8; ++r)
      dot += __shfl(acc[r], r, 32) + __shfl(acc[r], 24 + r, 32);

    if (lane == 0) partial[t] = termA * (float)nv - LAMBDA2 * dot;
  }
}

// Deterministic tree reduction of the 512 per-t partials into the scalar.
__global__ void __launch_bounds__(256)
vmf_reduce_kernel(const float* __restrict__ partial, float* __restrict__ out) {
  __shared__ float red[256];
  int tid  = threadIdx.x;
  red[tid] = partial[tid] + partial[tid + 256];
  __syncthreads();
#pragma unroll
  for (int s = 128; s > 0; s >>= 1) {
    if (tid < s) red[tid] += red[tid + s];
    __syncthreads();
  }
  if (tid == 0) out[0] = red[0];
}

extern "C" void kernel_launch(void* const* d_in, const int* in_sizes, int n_in,
                              void* d_out, int out_size, void* d_ws, size_t ws_size,
                              hipStream_t stream) {
  (void)in_sizes; (void)n_in; (void)out_size; (void)ws_size;
  const float* outputs   = (const float*)d_in[0];
  const int*   targets   = (const int*)d_in[1];
  const float* embedding = (const float*)d_in[2];
  float* partial = (float*)d_ws;   // T_DIM floats of scratch
  float* out     = (float*)d_out;  // scalar f32

  vmf_per_t_kernel<<<T_DIM, 256, 0, stream>>>(outputs, targets, embedding, partial);
  vmf_reduce_kernel<<<1, 256, 0, stream>>>(partial, out);
}